// DENlayer_6193342841090
// MI455X (gfx1250) — compile-verified
//
#include <hip/hip_runtime.h>
#include <hip/hip_bf16.h>

typedef __attribute__((ext_vector_type(2))) float v2f;
typedef __attribute__((ext_vector_type(8))) float v8f;
typedef unsigned int u32;
typedef __attribute__((ext_vector_type(4))) u32 u32x4;
typedef __attribute__((ext_vector_type(8))) u32 u32x8;

#define NE 128   // experts
#define NB 8192  // batch
#define ND 128   // feature dim
#define LDA_INV 129   // padded row stride for inversion LDS
#define LDB_S   132   // padded row stride for Sinv LDS in main kernel (128 + 4 pad)

// ---------------------------------------------------------------------------
// Kernel 1: batched in-place Gauss-Jordan inversion (no pivoting; Sigmas are
// 2*I + 0.05*noise -> strongly diagonally dominant, GJ is stable here).
// One workgroup (256 threads) per expert; matrix lives in LDS.
// ---------------------------------------------------------------------------
__global__ __launch_bounds__(256)
void den_invert_kernel(const float* __restrict__ Sigmas, float* __restrict__ Sinv) {
  extern __shared__ float lds[];
  float* M    = lds;                    // 128 x LDA_INV
  float* colk = lds + ND * LDA_INV;     // 128
  float* ps   = colk + ND;              // 1 (pivot inverse broadcast)

  const int e   = blockIdx.x;
  const int tid = threadIdx.x;
  const float* S = Sigmas + (size_t)e * ND * ND;

  for (int i = tid; i < ND * ND; i += 256) {
    int r = i >> 7, c = i & 127;
    M[r * LDA_INV + c] = S[i];
  }
  __syncthreads();

  for (int k = 0; k < ND; ++k) {
    if (tid == 0) ps[0] = 1.0f / M[k * LDA_INV + k];
    __syncthreads();
    const float pinv = ps[0];
    if (tid < ND) {
      // snapshot column k (pre-elimination), then scale row k
      colk[tid] = M[tid * LDA_INV + k];
      float v = M[k * LDA_INV + tid];
      M[k * LDA_INV + tid] = (tid == k) ? pinv : v * pinv;
    }
    __syncthreads();
    for (int idx = tid; idx < ND * ND; idx += 256) {
      int i = idx >> 7, j = idx & 127;
      if (i == k) continue;
      float f = colk[i];
      if (j == k) {
        M[i * LDA_INV + j] = -f * pinv;
      } else {
        M[i * LDA_INV + j] = M[i * LDA_INV + j] - f * M[k * LDA_INV + j];
      }
    }
    __syncthreads();
  }

  float* Dst = Sinv + (size_t)e * ND * ND;
  for (int i = tid; i < ND * ND; i += 256) {
    int r = i >> 7, c = i & 127;
    Dst[i] = M[r * LDA_INV + c];
  }
}

// ---------------------------------------------------------------------------
// TDM: DMA a 128x128 fp32 tile (row-major, stride 128) from global memory into
// LDS, inserting 4 DWORDs of padding after every 128 DWORDs (one row) so the
// LDS image has row stride LDB_S = 132 floats. One tensor_load_to_lds per
// block, issued by wave 0; TENSORcnt tracks completion.
// D# field packing per CDNA5 ISA.
// ---------------------------------------------------------------------------
__device__ static inline void tdm_load_sinv_to_lds(const float* gsrc, float* lds_dst) {
  const u32 lds_addr = (u32)(uintptr_t)lds_dst;  // LDS aperture: addr[31:0] = offset
  const unsigned long long ga = (unsigned long long)(uintptr_t)gsrc;

  u32x4 g0;
  g0[0] = 1u;                                            // count=1, user descriptor
  g0[1] = lds_addr;                                      // lds_addr (bits 63:32)
  g0[2] = (u32)ga;                                       // global_addr[31:0]
  g0[3] = (u32)((ga >> 32) & 0x1FFFFFFu) | (2u << 30);   // global_addr[56:32] | type=2

  u32x8 g1;
  g1[0] = (2u << 16)      // data_size = 2 -> 4 bytes
        | (1u << 20)      // pad_enable
        | (6u << 22)      // pad_interval = 6 -> every 128 DWORDs (one row)
        | (3u << 25);     // pad_amount   = 3 -> 4 DWORDs (16 B) of padding
  g1[1] = (128u << 16);   // tensor_dim0[15:0] = 128 (bits 63:48)
  g1[2] = (128u << 16);   // tensor_dim0[31:16]=0 | tensor_dim1[15:0] = 128
  g1[3] = (128u << 16);   // tensor_dim1[31:16]=0 | tile_dim0 = 128
  g1[4] = 128u;           // tile_dim1 = 128, tile_dim2 = 0
  g1[5] = 128u;           // tensor_dim0_stride[31:0] = 128 elements
  g1[6] = 0u;             // tensor_dim0_stride[47:32]=0 | tensor_dim1_stride lo = 0
  g1[7] = 0u;

  // Groups 2/3 omitted (NULL): 2-D tensor form.
  asm volatile("tensor_load_to_lds %0, %1" :: "s"(g0), "s"(g1) : "memory");
  __builtin_amdgcn_s_wait_tensorcnt(0);
}

// ---------------------------------------------------------------------------
// Kernel 2: mah[b,e] = dif_e[b,:] @ Sinv_e @ dif_e[b,:]^T via fp32 WMMA.
// Block = 256 threads = 8 waves; grid = (B/128, E).
// Sinv_e is staged into LDS by the Tensor Data Mover (one DMA per block).
// Each wave handles a 16-row tile of dif: T = dif16x128 @ Sinv128x128 using
// v_wmma_f32_16x16x4_f32 (32 K-steps per 16x16 output tile, 8 tiles), then
// mah = rowsum(T .* dif) with a 16-lane shfl_xor reduction.
// ---------------------------------------------------------------------------
__global__ __launch_bounds__(256)
void den_mahalanobis_kernel(const float* __restrict__ x,
                            const float* __restrict__ Centroids,
                            const float* __restrict__ Sinv,
                            float* __restrict__ out) {
  extern __shared__ float lds[];
  float* sinv = lds;                 // 128 x LDB_S (row-padded by the TDM)
  float* cen  = lds + ND * LDB_S;    // 128

  const int e    = blockIdx.y;
  const int tid  = threadIdx.x;
  const int lane = tid & 31;
  const int wave = tid >> 5;                 // 0..7
  const int b0   = blockIdx.x * 128 + wave * 16;

  // Wave 0 kicks the TDM for Sinv_e and waits on TENSORcnt; everyone else
  // just parks at the barrier. Centroid row is a trivial cooperative load.
  if (wave == 0) {
    tdm_load_sinv_to_lds(Sinv + (size_t)e * ND * ND, sinv);
  }
  if (tid < ND) cen[tid] = Centroids[(size_t)e * ND + tid];

  // A-fragment addressing (16x4 fp32 A layout):
  // lane 0-15 -> M = lane, K pair {0,1}; lane 16-31 -> M = lane-16, K pair {2,3}
  const int m     = lane & 15;
  const int half  = lane >> 4;       // 0 or 1
  const int nlane = lane & 15;       // N within C/B layout
  const int mc    = half ? 8 : 0;    // C-layout row offset for this half

  const float* xrow = x + (size_t)(b0 + m) * ND;
  __builtin_prefetch(xrow, 0, 1);    // global_prefetch_b8 on gfx1250

  __syncthreads();

  // Build 32 fp32 A-fragments for this wave's 16x128 dif tile.
  v2f a[32];
#pragma unroll
  for (int kk = 0; kk < 32; ++kk) {
    int k0 = 4 * kk + 2 * half;
    v2f t;
    t.x = xrow[k0]     - cen[k0];
    t.y = xrow[k0 + 1] - cen[k0 + 1];
    a[kk] = t;
  }

  float partial[8];
#pragma unroll
  for (int r = 0; r < 8; ++r) partial[r] = 0.0f;

#pragma unroll
  for (int j = 0; j < 8; ++j) {
    v8f acc = {};
#pragma unroll
    for (int kk = 0; kk < 32; ++kk) {
      int krow = 4 * kk + 2 * half;
      v2f bf;
      bf.x = sinv[krow * LDB_S + 16 * j + nlane];
      bf.y = sinv[(krow + 1) * LDB_S + 16 * j + nlane];
      acc = __builtin_amdgcn_wmma_f32_16x16x4_f32(
          /*neg_a=*/false, a[kk], /*neg_b=*/false, bf,
          /*c_mod=*/(short)0, acc, /*reuse_a=*/false, /*reuse_b=*/false);
    }
    // Element-wise multiply by dif in the C/D layout and accumulate per row.
    const int d = 16 * j + nlane;
    const float cd = cen[d];
#pragma unroll
    for (int r = 0; r < 8; ++r) {
      int mm = r + mc;
      float dv = x[(size_t)(b0 + mm) * ND + d] - cd;
      partial[r] += acc[r] * dv;
    }
  }

  // Reduce over the 16 lanes of each half (wave32; xor 1,2,4,8 stays in half).
#pragma unroll
  for (int r = 0; r < 8; ++r) {
    float v = partial[r];
    v += __shfl_xor(v, 1, 32);
    v += __shfl_xor(v, 2, 32);
    v += __shfl_xor(v, 4, 32);
    v += __shfl_xor(v, 8, 32);
    partial[r] = v;
  }

  if (nlane == 0) {
    // lane 0 holds rows b0+0..7 sums, lane 16 holds rows b0+8..15 sums
#pragma unroll
    for (int r = 0; r < 8; ++r) {
      out[(size_t)(b0 + r + mc) * NE + e] = partial[r];
    }
  }
}

extern "C" void kernel_launch(void* const* d_in, const int* in_sizes, int n_in,
                              void* d_out, int out_size, void* d_ws, size_t ws_size,
                              hipStream_t stream) {
  const float* x         = (const float*)d_in[0];  // [B, D]
  const float* Centroids = (const float*)d_in[1];  // [E, 1, D]
  const float* Sigmas    = (const float*)d_in[2];  // [E, D, D]
  float*       out       = (float*)d_out;          // [B, E]
  float*       SinvWs    = (float*)d_ws;           // [E, D, D] scratch (8 MB)

  const size_t ldsInv = (size_t)(ND * LDA_INV + ND + 4) * sizeof(float);   // ~66.6 KB
  const size_t ldsMah = (size_t)(ND * LDB_S + ND) * sizeof(float);         // ~68.1 KB

  den_invert_kernel<<<dim3(NE), dim3(256), ldsInv, stream>>>(Sigmas, SinvWs);
  den_mahalanobis_kernel<<<dim3(NB / 128, NE), dim3(256), ldsMah, stream>>>(
      x, Centroids, SinvWs, out);
}